// Downsample3d_35665408426075
// MI455X (gfx1250) — compile-verified
//
#include <hip/hip_runtime.h>

// ---------------------------------------------------------------------------
// Fused 3D FIR downsample (h,w,t each: 4-tap (1,3,3,1)/8, decimate by 2).
// features: (2,64,16,256,256) f32  ->  out: (2,64,8,128,128) f32
//
// Bandwidth-bound (~642 MB traffic, ~0.8 GFLOP => HBM roofline ~28 us):
// single fused pass, input staged global->LDS with CDNA5 async-to-LDS b128
// DMA (zero-padding chunks written directly, no separate memset pass), three
// barrier-separated separable FIR stages in LDS, one coalesced store.
// ---------------------------------------------------------------------------

#if defined(__HIP_DEVICE_COMPILE__) && \
    __has_builtin(__builtin_amdgcn_global_load_async_to_lds_b128) && \
    __has_builtin(__builtin_amdgcn_s_wait_asynccnt)
#define USE_ASYNC_LDS 1
#else
#define USE_ASYNC_LDS 0
#endif

// Builtin parameter types per hipcc diagnostic:
//   arg0: int __vector(4) addrspace(1)*   (global source, 16B)
//   arg1: int __vector(4) addrspace(3)*   (LDS destination, 16B)
typedef int v4i __attribute__((vector_size(4 * sizeof(int))));
typedef __attribute__((address_space(1))) v4i v4i_glb;
typedef __attribute__((address_space(3))) v4i v4i_lds;

// Tensor dims
#define T_IN   16
#define T_OUT  8
#define H_IN   256
#define W_IN   256
#define H_OUT  128
#define W_OUT  128

// Tile: one workgroup = one (n,c), 16x16 spatial output tile, all t.
#define HO_T   16
#define WO_T   16
#define ROWS   34          // 2*HO_T + 2 input rows needed
#define W0     40          // padded input tile width: 10 aligned float4 chunks
#define NCHUNK 10

// LDS layout (floats)
#define BUF0_F (T_IN * ROWS * W0)     // raw input tile      16*34*40 = 21760
#define BUFA_F (T_IN * ROWS * WO_T)   // after w-FIR         16*34*16 =  8704
#define BUFB_F (T_IN * HO_T * WO_T)   // after h-FIR         16*16*16 =  4096
#define LDS_FLOATS (BUF0_F + BUFA_F + BUFB_F)   // 34560 floats = 138240 B

extern "C" __global__ void __launch_bounds__(256)
downsample3d_fused(const float* __restrict__ x,
                   const float* __restrict__ filt,
                   float* __restrict__ out)
{
    extern __shared__ float lds[];
    float* buf0 = lds;
    float* bufA = lds + BUF0_F;
    float* bufB = bufA + BUFA_F;

    const int tid = threadIdx.x;
    const int bx  = blockIdx.x;          // 0 .. 2*64*8*8 - 1
    const int nc  = bx >> 6;             // (n*64 + c), 0..127
    const int ht  = (bx >> 3) & 7;
    const int wt  = bx & 7;
    const int ho0 = ht * HO_T;
    const int wo0 = wt * WO_T;
    const int hi0 = 2 * ho0 - 1;         // first needed input row
    const int a0  = 2 * wo0 - 4;         // 16B-aligned first tile col (wi0-3)

    // ---- tile fill: every 16B chunk owned by exactly one thread.
    //      valid chunk  -> async DMA global->LDS (b128)
    //      border chunk -> ds_store_b128 of zeros (h/w zero padding)
    const float* base = x + (size_t)nc * (T_IN * H_IN * W_IN);
    for (int id = tid; id < T_IN * ROWS * NCHUNK; id += 256) {
        const int t  = id / (ROWS * NCHUNK);
        const int rm = id % (ROWS * NCHUNK);
        const int r  = rm / NCHUNK;
        const int cw = rm % NCHUNK;
        const int hi = hi0 + r;
        const int c0 = a0 + 4 * cw;
        float* l = buf0 + (t * ROWS + r) * W0 + 4 * cw;
        if ((unsigned)hi < (unsigned)H_IN && (unsigned)c0 <= (unsigned)(W_IN - 4)) {
            const float* g = base + ((size_t)t * H_IN + hi) * W_IN + c0;
#if USE_ASYNC_LDS
            __builtin_amdgcn_global_load_async_to_lds_b128(
                (v4i_glb*)g, (v4i_lds*)l, 0, 0);
#else
            *(float4*)l = *(const float4*)g;
#endif
        } else {
            float4 z = {0.f, 0.f, 0.f, 0.f};
            *(float4*)l = z;
        }
    }

    const float f0 = filt[0], f1 = filt[1], f2 = filt[2], f3 = filt[3];

#if USE_ASYNC_LDS
    __builtin_amdgcn_s_wait_asynccnt(0);   // drain this wave's DMA ...
#endif
    __syncthreads();                        // ... then all waves' writes

    // ---- stage A: horizontal FIR + decimate (w).  In-tile start offset is
    //      always 3 because wi0 = 2*wo0-1 == a0+3.  16*34*16 items. ----
    for (int id = tid; id < T_IN * ROWS * WO_T; id += 256) {
        const int t  = id / (ROWS * WO_T);
        const int rm = id % (ROWS * WO_T);
        const int r  = rm / WO_T;
        const int wo = rm % WO_T;
        const float* row = buf0 + (t * ROWS + r) * W0 + 3 + 2 * wo;
        bufA[(t * ROWS + r) * WO_T + wo] =
            f0 * row[0] + f1 * row[1] + f2 * row[2] + f3 * row[3];
    }
    __syncthreads();

    // ---- stage B: vertical FIR + decimate (h).  16*16*16 items. ----
    for (int id = tid; id < T_IN * HO_T * WO_T; id += 256) {
        const int t  = id / (HO_T * WO_T);
        const int rm = id % (HO_T * WO_T);
        const int i  = rm / WO_T;
        const int wo = rm % WO_T;
        const float* col = bufA + (t * ROWS + 2 * i) * WO_T + wo;
        bufB[(t * HO_T + i) * WO_T + wo] =
            f0 * col[0] + f1 * col[WO_T] + f2 * col[2 * WO_T] + f3 * col[3 * WO_T];
    }
    __syncthreads();

    // ---- stage C: temporal FIR + decimate (t), store.  8*16*16 outputs. ----
    const int ho = tid >> 4;   // 0..15
    const int wo = tid & 15;   // 0..15
    const float fk[4] = {f0, f1, f2, f3};
    for (int to = 0; to < T_OUT; ++to) {
        float acc = 0.f;
        const int tb = 2 * to - 1;
#pragma unroll
        for (int k = 0; k < 4; ++k) {
            const int ti = tb + k;
            if ((unsigned)ti < (unsigned)T_IN)
                acc += fk[k] * bufB[(ti * HO_T + ho) * WO_T + wo];
        }
        const size_t oidx =
            (((size_t)nc * T_OUT + to) * H_OUT + (ho0 + ho)) * W_OUT + (wo0 + wo);
        out[oidx] = acc;
    }
}

extern "C" void kernel_launch(void* const* d_in, const int* in_sizes, int n_in,
                              void* d_out, int out_size, void* d_ws, size_t ws_size,
                              hipStream_t stream) {
    (void)in_sizes; (void)n_in; (void)d_ws; (void)ws_size; (void)out_size;
    const float* x    = (const float*)d_in[0];
    const float* filt = (const float*)d_in[1];
    float*       out  = (float*)d_out;

    const dim3 grid(2 * 64 * 8 * 8);   // (n*c) * h-tiles * w-tiles = 8192
    const dim3 block(256);
    const size_t shmem = (size_t)LDS_FLOATS * sizeof(float);
    hipLaunchKernelGGL(downsample3d_fused, grid, block, shmem, stream,
                       x, filt, out);
}